// Model_72773925863662
// MI455X (gfx1250) — compile-verified
//
#include <hip/hip_runtime.h>

typedef __attribute__((ext_vector_type(2))) float v2f;
typedef __attribute__((ext_vector_type(8))) float v8f;

#define H 64
#define TILE_M 16
#define WAVES_PER_BLK 4

// ---------------------------------------------------------------------------
// Utility: zero a float buffer
// ---------------------------------------------------------------------------
__global__ void k_zero(float* __restrict__ p, long n) {
    long i = (long)blockIdx.x * blockDim.x + threadIdx.x;
    if (i < n) p[i] = 0.0f;
}

// ---------------------------------------------------------------------------
// x_recipe0 = recipe_x @ lin_w + lin_b + recipe_emb   (K=10 tiny GEMM)
// one thread per output element
// ---------------------------------------------------------------------------
__global__ void k_init_recipe(const float* __restrict__ rx,
                              const float* __restrict__ lw,
                              const float* __restrict__ lb,
                              const float* __restrict__ remb,
                              float* __restrict__ xr, long n) {
    long i = (long)blockIdx.x * blockDim.x + threadIdx.x;
    if (i >= n) return;
    long r = i >> 6;
    int  h = (int)(i & 63);
    float acc = lb[h] + remb[i];
    const float* row = rx + r * 10;
#pragma unroll
    for (int k = 0; k < 10; ++k) acc += row[k] * lw[k * H + h];
    xr[i] = acc;
}

// ---------------------------------------------------------------------------
// Degree histograms (one thread per edge)
// ---------------------------------------------------------------------------
__global__ void k_degree(const int* __restrict__ src, const int* __restrict__ dst,
                         float* __restrict__ degu, float* __restrict__ degr, int E) {
    int e = blockIdx.x * blockDim.x + threadIdx.x;
    if (e >= E) return;
    atomicAdd(&degu[src[e]], 1.0f);
    atomicAdd(&degr[dst[e]], 1.0f);
}

__global__ void k_invert(float* __restrict__ d, long n) {
    long i = (long)blockIdx.x * blockDim.x + threadIdx.x;
    if (i < n) d[i] = 1.0f / fmaxf(d[i], 1.0f);
}

// ---------------------------------------------------------------------------
// agg[sc[e]] += x[gt[e]]  : one wave (32 lanes) per edge, 2 floats per lane
// ---------------------------------------------------------------------------
__global__ void k_scatter_add(const int* __restrict__ gt, const int* __restrict__ sc,
                              const float* __restrict__ x, float* __restrict__ agg, int E) {
    int wid  = (int)(((long)blockIdx.x * blockDim.x + threadIdx.x) >> 5);
    int lane = threadIdx.x & 31;
    if (wid >= E) return;
    long g = gt[wid];
    long s = sc[wid];
    const float* xs = x + g * H;
    float* ag = agg + s * H;
    atomicAdd(&ag[lane],      xs[lane]);
    atomicAdd(&ag[lane + 32], xs[lane + 32]);
}

// ---------------------------------------------------------------------------
// Fused SAGE projection via WMMA f32 16x16x4:
//   out = maybe_relu( (agg*invdeg) @ Wl + bl + xdst @ Wr )
// Combined GEMM: A = [agg*invdeg | xdst]  (M x 128),  B = [Wl ; Wr] (128 x 64)
// B is staged transposed in LDS ([n][k], stride 132) so each lane's (k,k+1)
// fragment is one ds_load_b64. A fragments are 8-byte global loads
// (f32 A layout: lane l -> m = l%16, kbase = k0 + 2*(l/16)).
// ---------------------------------------------------------------------------
__device__ __forceinline__ v8f wmma4(v2f a, v2f b, v8f c) {
    return __builtin_amdgcn_wmma_f32_16x16x4_f32(
        /*neg_a=*/false, a, /*neg_b=*/false, b,
        /*c_mod=*/(short)0, c, /*reuse_a=*/false, /*reuse_b=*/false);
}

#define BSTRIDE 132

__global__ __launch_bounds__(WAVES_PER_BLK * 32)
void k_sage_gemm(const float* __restrict__ agg, const float* __restrict__ invdeg,
                 const float* __restrict__ xdst,
                 const float* __restrict__ Wlp, const float* __restrict__ Wrp,
                 const float* __restrict__ blp,
                 float* __restrict__ out, int nrows, int relu) {
    __shared__ float BsT[H * BSTRIDE];   // [n][k] transposed, k = 0..127

    // Stage B = [Wl ; Wr] transposed into LDS (coalesced global reads)
    for (int i = threadIdx.x; i < 128 * H; i += WAVES_PER_BLK * 32) {
        int k = i >> 6;
        int n = i & 63;
        float v = (k < 64) ? Wlp[k * H + n] : Wrp[(k - 64) * H + n];
        BsT[n * BSTRIDE + k] = v;
    }
    __syncthreads();

    int wave = threadIdx.x >> 5;
    int lane = threadIdx.x & 31;
    long tile = (long)blockIdx.x * WAVES_PER_BLK + wave;
    long m0 = tile * TILE_M;
    if (m0 >= nrows) return;

    int  mlo   = lane & 15;          // row within tile for A / col within ntile for B,C
    int  khalf = lane >> 4;          // 0: k0,k0+1   1: k0+2,k0+3
    long m     = m0 + mlo;
    float idg  = invdeg[m];
    const float* A1 = agg  + m * H;  // neighbor-mean part (scaled by invdeg)
    const float* A2 = xdst + m * H;  // root part

    v8f acc[4];
    acc[0] = {}; acc[1] = {}; acc[2] = {}; acc[3] = {};

    // K = 0..63  : mean @ Wl
#pragma unroll
    for (int k0 = 0; k0 < 64; k0 += 4) {
        int kb = k0 + 2 * khalf;
        v2f a = *(const v2f*)&A1[kb];
        a.x *= idg; a.y *= idg;
#pragma unroll
        for (int t = 0; t < 4; ++t) {
            v2f b = *(const v2f*)&BsT[(t * 16 + mlo) * BSTRIDE + kb];
            acc[t] = wmma4(a, b, acc[t]);
        }
    }
    // K = 64..127 : xdst @ Wr
#pragma unroll
    for (int k0 = 0; k0 < 64; k0 += 4) {
        int kb = k0 + 2 * khalf;
        v2f a = *(const v2f*)&A2[kb];
#pragma unroll
        for (int t = 0; t < 4; ++t) {
            v2f b = *(const v2f*)&BsT[(t * 16 + mlo) * BSTRIDE + 64 + kb];
            acc[t] = wmma4(a, b, acc[t]);
        }
    }

    // Epilogue: C layout — VGPR j: lanes 0-15 -> M=j, N=lane; lanes 16-31 -> M=8+j
    long mbase = m0 + (long)(khalf << 3);
#pragma unroll
    for (int t = 0; t < 4; ++t) {
        int gn = t * 16 + mlo;
        float bv = blp[gn];
#pragma unroll
        for (int j = 0; j < 8; ++j) {
            float v = acc[t][j] + bv;
            if (relu) v = fmaxf(v, 0.0f);
            out[(mbase + j) * H + gn] = v;
        }
    }
}

// ---------------------------------------------------------------------------
// Classifier: out[i] = dot(xu[e0[i]], xr[e1[i]]) — one wave per label edge
// ---------------------------------------------------------------------------
__global__ void k_classifier(const float* __restrict__ xu, const float* __restrict__ xr,
                             const int* __restrict__ e0, const int* __restrict__ e1,
                             float* __restrict__ out, int L) {
    int wid  = (int)(((long)blockIdx.x * blockDim.x + threadIdx.x) >> 5);
    int lane = threadIdx.x & 31;
    if (wid >= L) return;
    const float* a = xu + (long)e0[wid] * H;
    const float* b = xr + (long)e1[wid] * H;
    float s = a[lane] * b[lane] + a[lane + 32] * b[lane + 32];
#pragma unroll
    for (int off = 16; off > 0; off >>= 1) s += __shfl_xor(s, off, 32);
    if (lane == 0) out[wid] = s;
}

// ---------------------------------------------------------------------------
// Launch orchestration
// ---------------------------------------------------------------------------
extern "C" void kernel_launch(void* const* d_in, const int* in_sizes, int n_in,
                              void* d_out, int out_size, void* d_ws, size_t ws_size,
                              hipStream_t stream) {
    const float* recipe_x = (const float*)d_in[2];
    const int*   eidx     = (const int*)d_in[3];
    const int*   elidx    = (const int*)d_in[4];
    const float* user_emb = (const float*)d_in[5];
    const float* rec_emb  = (const float*)d_in[6];
    const float* lin_w    = (const float*)d_in[7];
    const float* lin_b    = (const float*)d_in[8];
    const float* Wl       = (const float*)d_in[9];
    const float* bl       = (const float*)d_in[10];
    const float* Wr       = (const float*)d_in[11];

    const int E  = in_sizes[3] / 2;
    const int L  = in_sizes[4] / 2;
    const int NU = in_sizes[5] / H;
    const int NR = in_sizes[6] / H;
    const int* src = eidx;          // user endpoint
    const int* dst = eidx + E;      // recipe endpoint
    const int* el0 = elidx;         // user side of label edge
    const int* el1 = elidx + L;     // recipe side of label edge

    const long FEATR = (long)NR * H;
    const long FEATU = (long)NU * H;

    // Workspace carve-out (~258 MB)
    float* XR_A = (float*)d_ws;            // recipe feats (init / layer1 out)
    float* XR_B = XR_A + FEATR;            // recipe feats layer0 out
    float* XU_B = XR_B + FEATR;            // user feats layer0 out
    float* XU_C = XU_B + FEATU;            // user feats layer1 out
    float* AGG  = XU_C + FEATU;            // shared aggregation buffer
    float* DEGU = AGG  + ((FEATR > FEATU) ? FEATR : FEATU);
    float* DEGR = DEGU + NU;               // contiguous with DEGU

    const int T = 256;
    auto blocks = [](long n, int t) { return (unsigned)((n + t - 1) / t); };

    // Degrees (edge-structure only): zero, count, invert
    hipLaunchKernelGGL(k_zero, dim3(blocks(NU + NR, T)), dim3(T), 0, stream, DEGU, (long)(NU + NR));
    hipLaunchKernelGGL(k_degree, dim3(blocks(E, T)), dim3(T), 0, stream, src, dst, DEGU, DEGR, E);
    hipLaunchKernelGGL(k_invert, dim3(blocks(NU + NR, T)), dim3(T), 0, stream, DEGU, (long)(NU + NR));

    // Initial recipe features
    hipLaunchKernelGGL(k_init_recipe, dim3(blocks(FEATR, T)), dim3(T), 0, stream,
                       recipe_x, lin_w, lin_b, rec_emb, XR_A, FEATR);

    const unsigned scatterBlocks = blocks((long)E * 32, T);
    const unsigned gemmBlocksR = (unsigned)((NR / TILE_M + WAVES_PER_BLK - 1) / WAVES_PER_BLK);
    const unsigned gemmBlocksU = (unsigned)((NU / TILE_M + WAVES_PER_BLK - 1) / WAVES_PER_BLK);
    auto WL = [&](int layer, int rel) { return Wl + (size_t)(layer * 2 + rel) * H * H; };
    auto WR = [&](int layer, int rel) { return Wr + (size_t)(layer * 2 + rel) * H * H; };
    auto BL = [&](int layer, int rel) { return bl + (size_t)(layer * 2 + rel) * H; };

    const float* xu_in = user_emb;
    const float* xr_in = XR_A;
    float* xr_out = XR_B;
    float* xu_out = XU_B;

    for (int layer = 0; layer < 2; ++layer) {
        int relu = (layer == 0) ? 1 : 0;

        // new_recipe: agg[dst] += xu_in[src]; project
        hipLaunchKernelGGL(k_zero, dim3(blocks(FEATR, T)), dim3(T), 0, stream, AGG, FEATR);
        hipLaunchKernelGGL(k_scatter_add, dim3(scatterBlocks), dim3(T), 0, stream,
                           src, dst, xu_in, AGG, E);
        hipLaunchKernelGGL(k_sage_gemm, dim3(gemmBlocksR), dim3(WAVES_PER_BLK * 32), 0, stream,
                           AGG, DEGR, xr_in, WL(layer, 0), WR(layer, 0), BL(layer, 0),
                           xr_out, NR, relu);

        // new_user: agg[src] += xr_in[dst]; project  (uses OLD recipe feats)
        hipLaunchKernelGGL(k_zero, dim3(blocks(FEATU, T)), dim3(T), 0, stream, AGG, FEATU);
        hipLaunchKernelGGL(k_scatter_add, dim3(scatterBlocks), dim3(T), 0, stream,
                           dst, src, xr_in, AGG, E);
        hipLaunchKernelGGL(k_sage_gemm, dim3(gemmBlocksU), dim3(WAVES_PER_BLK * 32), 0, stream,
                           AGG, DEGU, xu_in, WL(layer, 1), WR(layer, 1), BL(layer, 1),
                           xu_out, NU, relu);

        // rotate ping-pong: layer1 writes recipe into XR_A (free), user into XU_C
        xr_in = xr_out; xu_in = xu_out;
        xr_out = XR_A;  xu_out = XU_C;
    }

    // Classifier over label edges
    hipLaunchKernelGGL(k_classifier, dim3(blocks((long)L * 32, T)), dim3(T), 0, stream,
                       xu_in, xr_in, el0, el1, (float*)d_out, L);
}